// GCNNet_24824910970942
// MI455X (gfx1250) — compile-verified
//
#include <hip/hip_runtime.h>
#include <hip/hip_bf16.h>

#define N_NODES 100000
#define N_EDGES 1600000
#define F_IN    128
#define F_HID   64
#define N_CLS   16

typedef float v2f __attribute__((ext_vector_type(2)));
typedef float v8f __attribute__((ext_vector_type(8)));

// ---------------- degree / normalization ----------------

__global__ void k_fill_one(float* __restrict__ deg, int n) {
    int i = blockIdx.x * blockDim.x + threadIdx.x;
    if (i < n) deg[i] = 1.0f;                    // self-loop contributes 1
}

__global__ void k_deg_edges(float* __restrict__ deg, const int* __restrict__ dst, int e) {
    int i = blockIdx.x * blockDim.x + threadIdx.x;
    if (i < e) atomicAdd(&deg[dst[i]], 1.0f);
}

__global__ void k_rsqrt(float* __restrict__ deg, int n) {
    int i = blockIdx.x * blockDim.x + threadIdx.x;
    if (i < n) deg[i] = __frsqrt_rn(deg[i]);     // deg >= 1 always (self-loops)
}

// ---------------- dense GEMM via V_WMMA_F32_16X16X4_F32 ----------------
// out[M,Fout] = (RELU ? relu(A) : A)[M,K] @ W[K,Fout]  (+ bias[n] if BIAS)
// One wave per 16x16 output tile; K templated so the loop fully unrolls and
// every load becomes a constant-immediate-offset global_load.

template<int K, int Fout, bool RELU, bool BIAS>
__global__ __launch_bounds__(256)
void k_gemm_wmma(const float* __restrict__ A, const float* __restrict__ W,
                 float* __restrict__ out, int M, const float* __restrict__ bias) {
    const int lane  = threadIdx.x & 31;
    const int wave  = threadIdx.x >> 5;
    const int wpb   = blockDim.x >> 5;
    const int tile  = blockIdx.x * wpb + wave;
    constexpr int nTn = Fout >> 4;          // N-tiles
    const int mt    = tile / nTn;
    const int nt    = tile - mt * nTn;
    if (mt >= (M >> 4)) return;             // wave-uniform guard: EXEC stays all-ones
    const int m0    = mt << 4;
    const int n0    = nt << 4;
    const int half  = lane >> 4;            // 0: lanes 0-15, 1: lanes 16-31
    const int l16   = lane & 15;

    v8f c;
    if (BIAS) {
        const float bv = bias[n0 + l16];
        #pragma unroll
        for (int r = 0; r < 8; ++r) c[r] = bv;
    } else {
        #pragma unroll
        for (int r = 0; r < 8; ++r) c[r] = 0.0f;
    }

    const float* arow = A + (size_t)(m0 + l16) * K + (half << 1);  // lane's A row
    const float* wcol = W + (size_t)(half << 1) * Fout + n0 + l16; // lane's B column

    #pragma unroll
    for (int k0 = 0; k0 < K; k0 += 4) {
        float a0 = arow[k0];
        float a1 = arow[k0 + 1];
        if (RELU) {
            a0 = (a0 < 0.0f) ? 0.0f : a0;
            a1 = (a1 < 0.0f) ? 0.0f : a1;
        }
        v2f av; av[0] = a0; av[1] = a1;
        v2f bv;
        bv[0] = wcol[(size_t)k0 * Fout];
        bv[1] = wcol[(size_t)(k0 + 1) * Fout];
        c = __builtin_amdgcn_wmma_f32_16x16x4_f32(
                /*neg_a=*/false, av, /*neg_b=*/false, bv,
                /*c_mod=*/(short)0, c, /*reuse_a=*/false, /*reuse_b=*/false);
    }

    // D layout: VGPR r -> row m0 + r + half*8, col n0 + l16
    float* orow = out + (size_t)(m0 + (half << 3)) * Fout + n0 + l16;
    #pragma unroll
    for (int r = 0; r < 8; ++r) orow[(size_t)r * Fout] = c[r];
}

// ---------------- aggregation: init with bias + self-loop ----------------
// out[i][f] = b[f] + dinv[i]^2 * h[i][f]   (F fixed 64; float4-vectorized)

__global__ void k_init_agg(const float4* __restrict__ h, float4* __restrict__ out,
                           const float* __restrict__ dinv, const float4* __restrict__ b,
                           int n) {
    int idx = blockIdx.x * blockDim.x + threadIdx.x;
    if (idx >= n * (F_HID / 4)) return;
    int i  = idx >> 4;          // node
    int q  = idx & 15;          // float4 chunk within row
    float di = dinv[i];
    float w  = di * di;
    float4 hv = h[idx];
    float4 bv = b[q];
    float4 o;
    o.x = bv.x + w * hv.x;
    o.y = bv.y + w * hv.y;
    o.z = bv.z + w * hv.z;
    o.w = bv.w + w * hv.w;
    out[idx] = o;
}

// ---------------- aggregation: edge scatter with f32 atomics ----------------
// out[dst[e]][f] += dinv[src]*dinv[dst] * h[src[e]][f]
// One float4 of payload per thread: 16 threads per edge, b128 gather + 4 atomics.

__global__ void k_scatter(const float* __restrict__ h, float* __restrict__ out,
                          const int* __restrict__ src, const int* __restrict__ dst,
                          const float* __restrict__ dinv, int e) {
    int idx = blockIdx.x * blockDim.x + threadIdx.x;
    if (idx >= e * (F_HID / 4)) return;      // 25.6M < 2^31
    int ed = idx >> 4;
    int q  = idx & 15;
    int s = src[ed];
    int d = dst[ed];
    float w = dinv[s] * dinv[d];
    const float4 hv = ((const float4*)(h + (size_t)s * F_HID))[q];
    float* o = out + (size_t)d * F_HID + (q << 2);
    atomicAdd(o + 0, w * hv.x);
    atomicAdd(o + 1, w * hv.y);
    atomicAdd(o + 2, w * hv.z);
    atomicAdd(o + 3, w * hv.w);
}

// ---------------- host-side orchestration ----------------

extern "C" void kernel_launch(void* const* d_in, const int* in_sizes, int n_in,
                              void* d_out, int out_size, void* d_ws, size_t ws_size,
                              hipStream_t stream) {
    const float* x   = (const float*)d_in[0];
    const int*   ei  = (const int*)  d_in[1];
    const float* W0  = (const float*)d_in[2];
    const float* b0  = (const float*)d_in[3];
    const float* W1  = (const float*)d_in[4];
    const float* b1  = (const float*)d_in[5];
    const float* W2  = (const float*)d_in[6];
    const float* b2  = (const float*)d_in[7];
    const float* Wc  = (const float*)d_in[8];
    const float* bc  = (const float*)d_in[9];
    float* out = (float*)d_out;

    const int* src = ei;
    const int* dst = ei + N_EDGES;

    // workspace: dinv (padded for alignment) | hA (N x 64) | hB (N x 64)
    float* dinv = (float*)d_ws;
    float* hA   = dinv + 102400;                     // 256B-aligned region
    float* hB   = hA + (size_t)N_NODES * F_HID;

    const int T = 256;
    const int gN   = (N_NODES + T - 1) / T;
    const int gE   = (N_EDGES + T - 1) / T;
    const int gNF4 = (N_NODES * (F_HID / 4) + T - 1) / T;     // 1.6M threads
    const int gEF4 = (int)(((size_t)N_EDGES * (F_HID / 4) + T - 1) / T);  // 25.6M threads
    const int tilesHid = (N_NODES / 16) * (F_HID / 16);  // 25000
    const int tilesCls = (N_NODES / 16) * (N_CLS / 16);  // 6250
    const int gGH = (tilesHid + 7) / 8;                  // 8 waves / 256-thread block
    const int gGC = (tilesCls + 7) / 8;

    // normalization
    k_fill_one <<<gN, T, 0, stream>>>(dinv, N_NODES);
    k_deg_edges<<<gE, T, 0, stream>>>(dinv, dst, N_EDGES);
    k_rsqrt    <<<gN, T, 0, stream>>>(dinv, N_NODES);

    // layer 0: hA = x @ W0 ; hB = b0 + dinv^2*hA + scatter(hA)
    k_gemm_wmma<F_IN, F_HID, false, false><<<gGH, T, 0, stream>>>(x, W0, hA, N_NODES, nullptr);
    k_init_agg <<<gNF4, T, 0, stream>>>((const float4*)hA, (float4*)hB, dinv, (const float4*)b0, N_NODES);
    k_scatter  <<<gEF4, T, 0, stream>>>(hA, hB, src, dst, dinv, N_EDGES);

    // layer 1 (relu fused into GEMM A-load)
    k_gemm_wmma<F_HID, F_HID, true, false><<<gGH, T, 0, stream>>>(hB, W1, hA, N_NODES, nullptr);
    k_init_agg <<<gNF4, T, 0, stream>>>((const float4*)hA, (float4*)hB, dinv, (const float4*)b1, N_NODES);
    k_scatter  <<<gEF4, T, 0, stream>>>(hA, hB, src, dst, dinv, N_EDGES);

    // layer 2
    k_gemm_wmma<F_HID, F_HID, true, false><<<gGH, T, 0, stream>>>(hB, W2, hA, N_NODES, nullptr);
    k_init_agg <<<gNF4, T, 0, stream>>>((const float4*)hA, (float4*)hB, dinv, (const float4*)b2, N_NODES);
    k_scatter  <<<gEF4, T, 0, stream>>>(hA, hB, src, dst, dinv, N_EDGES);

    // classifier: logits = hB @ Wc + bc   (no relu after layer 2 in reference)
    k_gemm_wmma<F_HID, N_CLS, false, true><<<gGC, T, 0, stream>>>(hB, Wc, out, N_NODES, bc);
}